// FPSSampler_7876970020952
// MI455X (gfx1250) — compile-verified
//
#include <hip/hip_runtime.h>

// FPS: B=8 batches, P=65536 points each, select M=4096 per batch.
// Input rows are (b, x, y, z) float32, 16B each. Batch column already sorted,
// argsort is stable -> identity permutation -> FPS on contiguous chunks.

#define B_      8
#define P_      65536
#define M_      4096
#define THREADS 1024
#define NPT     (P_ / THREADS)   // 64 points per thread, dist kept in VGPRs
#define NWAVES  (THREADS / 32)   // 32 waves (wave32)

typedef __attribute__((ext_vector_type(4))) int v4i;
typedef __attribute__((address_space(1))) v4i* v4i_gptr;  // global
typedef __attribute__((address_space(3))) v4i* v4i_lptr;  // LDS

// Broadcast one 16B global row into LDS. Uses the CDNA5 async global->LDS
// path when the builtin exists (gfx1250: global_load_async_to_lds_b128 +
// s_wait_asynccnt), otherwise a plain load/store.
__device__ __forceinline__ void fetch_row_to_lds(const float4* __restrict__ src,
                                                 float4* dst_lds) {
#if defined(__gfx1250__) && __has_builtin(__builtin_amdgcn_global_load_async_to_lds_b128) && __has_builtin(__builtin_amdgcn_s_wait_asynccnt)
    v4i* sv = (v4i*)src;                     // generic, non-const
    v4i* dv = (v4i*)dst_lds;                 // generic
    __builtin_amdgcn_global_load_async_to_lds_b128(
        (v4i_gptr)sv,                        // addrspacecast generic -> global
        (v4i_lptr)dv,                        // addrspacecast generic -> LDS
        /*imm offset*/ 0, /*cpol*/ 0);
    __builtin_amdgcn_s_wait_asynccnt(0);
#else
    *dst_lds = *src;
#endif
}

__global__ __launch_bounds__(THREADS) void
fps_kernel(const float* __restrict__ in, float* __restrict__ out) {
    const int b    = blockIdx.x;        // one workgroup per batch
    const int tid  = threadIdx.x;
    const int lane = tid & 31;
    const int wave = tid >> 5;

    const float4* __restrict__ rows = (const float4*)in + (size_t)b * P_;
    float4* __restrict__ orow       = (float4*)out + (size_t)b * M_;

    __shared__ float4 s_q;              // current query row (b,x,y,z)
    __shared__ float  s_rd[NWAVES];     // per-wave best dist
    __shared__ int    s_ri[NWAVES];     // per-wave best index

    // dist array lives entirely in VGPRs (loops over it are fully unrolled)
    float dist[NPT];
#pragma unroll
    for (int j = 0; j < NPT; ++j) dist[j] = 1e10f;

    // Warm the cache hierarchy for this batch's stream (global_prefetch_b8).
#pragma unroll
    for (int j = 0; j < NPT; ++j)
        __builtin_prefetch((const void*)(rows + j * THREADS + tid), 0, 3);

    if (tid == 0) {
        fetch_row_to_lds(rows + 0, &s_q);   // query 0 = point 0
        orow[0] = s_q;                      // emit first sample
    }
    __syncthreads();

    for (int i = 1; i < M_; ++i) {
        const float qx = s_q.y, qy = s_q.z, qz = s_q.w;

        float bestD = -1.0f;
        int   bestI = 0;
        // Full unroll: 64 coalesced b128 loads + FP32 min-update + argmax.
#pragma unroll
        for (int j = 0; j < NPT; ++j) {
            const float4 r = rows[j * THREADS + tid];
            const float dx = r.y - qx;
            const float dy = r.z - qy;
            const float dz = r.w - qz;
            const float d  = dx * dx + dy * dy + dz * dz;
            const float nd = fminf(dist[j], d);
            dist[j] = nd;
            // strictly-greater keeps smaller j on ties (index grows with j)
            if (nd > bestD) { bestD = nd; bestI = j * THREADS + tid; }
        }

        // wave32 argmax reduction, first-occurrence tie-break (smaller index)
#pragma unroll
        for (int off = 16; off > 0; off >>= 1) {
            const float od = __shfl_xor(bestD, off, 32);
            const int   oi = __shfl_xor(bestI, off, 32);
            if (od > bestD || (od == bestD && oi < bestI)) { bestD = od; bestI = oi; }
        }
        if (lane == 0) { s_rd[wave] = bestD; s_ri[wave] = bestI; }
        __syncthreads();

        // cross-wave reduction: wave 0 reduces the 32 per-wave candidates
        if (wave == 0) {
            float d  = s_rd[lane];
            int   ix = s_ri[lane];
#pragma unroll
            for (int off = 16; off > 0; off >>= 1) {
                const float od = __shfl_xor(d, off, 32);
                const int   oi = __shfl_xor(ix, off, 32);
                if (od > d || (od == d && oi < ix)) { d = od; ix = oi; }
            }
            if (lane == 0) {
                // async-broadcast the winner row into LDS, then emit it
                fetch_row_to_lds(rows + ix, &s_q);
                orow[i] = s_q;
            }
        }
        __syncthreads();   // everyone sees the new query in s_q
    }
}

extern "C" void kernel_launch(void* const* d_in, const int* in_sizes, int n_in,
                              void* d_out, int out_size, void* d_ws, size_t ws_size,
                              hipStream_t stream) {
    (void)in_sizes; (void)n_in; (void)d_ws; (void)ws_size; (void)out_size;
    const float* in = (const float*)d_in[0];   // [B*P, 4] f32
    float* out      = (float*)d_out;           // [B*M, 4] f32
    fps_kernel<<<dim3(B_), dim3(THREADS), 0, stream>>>(in, out);
}